// DiceLoss2_47158740910860
// MI455X (gfx1250) — compile-verified
//
#include <hip/hip_runtime.h>

typedef __attribute__((ext_vector_type(2))) float v2f;
typedef __attribute__((ext_vector_type(4))) float v4f;
typedef __attribute__((ext_vector_type(8))) float v8f;

#define NB 16
#define NC 21
#define HW (512 * 512)
#define NPLANES (NB * NC)
#define SPLIT 8
#define CHUNK (HW / SPLIT)      // 32768 elements per block
#define THREADS 256
#define WAVES (THREADS / 32)
#define PER_ITER (THREADS * 4)  // 1024 elements per block-iteration

// Pass 1: per-(plane, split) partial reductions of ta, tb, inter.
// ta/tb ride the f32 WMMA pipe via the ones-matrix trick:
//   trace(A_yt x ones(4x16)) == sum of all 64 A elements (layout-agnostic)
//   trace(ones(16x4) x B_yp) == sum of all 64 B elements (layout-agnostic)
// inter uses 4 independent VALU FMA chains (exact element pairing),
// co-executing with the XDL WMMA ops.
__global__ __launch_bounds__(THREADS) void dice_partials_kernel(
    const float* __restrict__ y_pred, const float* __restrict__ y_true,
    float* __restrict__ partials)
{
    const int plane = blockIdx.x;   // b*NC + c, matches (B,C,H,W) layout
    const int split = blockIdx.y;
    const int lane  = threadIdx.x & 31;
    const int wave  = threadIdx.x >> 5;

    const size_t base = (size_t)plane * HW + (size_t)split * CHUNK
                      + (size_t)wave * 128 + (size_t)lane * 4;
    const float* pt = y_true + base;
    const float* pp = y_pred + base;

    v8f c_ta = {};          // accumulates yt sums (read out via trace)
    v8f c_tb = {};          // accumulates yp sums (read out via trace)
    float i0 = 0.0f, i1 = 0.0f, i2 = 0.0f, i3 = 0.0f;  // 4 independent chains
    const v2f ones = {1.0f, 1.0f};

#pragma unroll 4
    for (int i = 0; i < CHUNK; i += PER_ITER) {
        v4f a4 = *(const v4f*)(pt + i);   // global_load_b128
        v4f b4 = *(const v4f*)(pp + i);
        v2f alo = {a4.x, a4.y}, ahi = {a4.z, a4.w};
        v2f blo = {b4.x, b4.y}, bhi = {b4.z, b4.w};

        // (neg_a, A, neg_b, B, c_mod, C, reuse_a, reuse_b)
        c_ta = __builtin_amdgcn_wmma_f32_16x16x4_f32(
            false, alo, false, ones, (short)0, c_ta, false, false);
        c_ta = __builtin_amdgcn_wmma_f32_16x16x4_f32(
            false, ahi, false, ones, (short)0, c_ta, false, false);
        c_tb = __builtin_amdgcn_wmma_f32_16x16x4_f32(
            false, ones, false, blo, (short)0, c_tb, false, false);
        c_tb = __builtin_amdgcn_wmma_f32_16x16x4_f32(
            false, ones, false, bhi, (short)0, c_tb, false, false);

        i0 = fmaf(a4.x, b4.x, i0);
        i1 = fmaf(a4.y, b4.y, i1);
        i2 = fmaf(a4.z, b4.z, i2);
        i3 = fmaf(a4.w, b4.w, i3);
    }

    float inter = (i0 + i1) + (i2 + i3);   // fixed order -> deterministic

    // Trace extraction: D[m][n] lives at lane (n + 16*(m>=8)), vgpr (m & 7).
    const int n     = lane & 15;
    const int mbase = (lane >> 4) << 3;    // 0 or 8
    float tta = 0.0f, ttb = 0.0f;
#pragma unroll
    for (int r = 0; r < 8; ++r) {
        const bool diag = (mbase + r) == n;
        tta += diag ? c_ta[r] : 0.0f;
        ttb += diag ? c_tb[r] : 0.0f;
    }

    // Wave reduction (wave32).
    for (int off = 16; off > 0; off >>= 1) {
        tta   += __shfl_xor(tta, off, 32);
        ttb   += __shfl_xor(ttb, off, 32);
        inter += __shfl_xor(inter, off, 32);
    }

    __shared__ float smem[WAVES][3];
    if (lane == 0) {
        smem[wave][0] = tta;
        smem[wave][1] = ttb;
        smem[wave][2] = inter;
    }
    __syncthreads();
    if (threadIdx.x == 0) {
        float s0 = 0.0f, s1 = 0.0f, s2 = 0.0f;
        for (int w = 0; w < WAVES; ++w) {   // fixed order -> deterministic
            s0 += smem[w][0];
            s1 += smem[w][1];
            s2 += smem[w][2];
        }
        float* out = partials + ((size_t)plane * SPLIT + split) * 3;
        out[0] = s0;
        out[1] = s1;
        out[2] = s2;
    }
}

// Pass 2: single block, fixed-order reductions -> deterministic scalar loss.
__global__ __launch_bounds__(384) void dice_finalize_kernel(
    const float* __restrict__ partials, const int* __restrict__ bg_ptr,
    float* __restrict__ out)
{
    __shared__ float dice_s[NPLANES];
    __shared__ float valid_s[NPLANES];
    __shared__ float tmp_s[NB];
    __shared__ float bv_s[NB];

    const int t  = threadIdx.x;
    const int bg = *bg_ptr;

    if (t < NPLANES) {
        const int c = t % NC;
        float ta = 0.0f, tb = 0.0f, inter = 0.0f;
        for (int s = 0; s < SPLIT; ++s) {
            const float* p = partials + ((size_t)t * SPLIT + s) * 3;
            ta    += p[0];
            tb    += p[1];
            inter += p[2];
        }
        const bool incl  = (c >= bg);
        const bool valid = incl && (ta != 0.0f);
        const float dice = valid ? (2.0f * inter / (ta + tb + 1e-11f)) : 0.0f;
        dice_s[t]  = dice;                 // already 0 when !valid
        valid_s[t] = valid ? 1.0f : 0.0f;
    }
    __syncthreads();

    if (t < NB) {
        float dsum = 0.0f, cpt2 = 0.0f;
        for (int c = 0; c < NC; ++c) {
            dsum += dice_s[t * NC + c];
            cpt2 += valid_s[t * NC + c];
        }
        const float denom   = cpt2 - (float)bg;
        const bool  bvalid  = (denom != 0.0f);
        tmp_s[t] = bvalid ? (dsum / denom) : 0.0f;
        bv_s[t]  = bvalid ? 1.0f : 0.0f;
    }
    __syncthreads();

    if (t == 0) {
        float s = 0.0f, cpt1 = 0.0f;
        for (int b = 0; b < NB; ++b) {
            s    += tmp_s[b];
            cpt1 += bv_s[b];
        }
        const float loss = 1.0f - s / fmaxf(cpt1, 1.0f);
        out[0] = (cpt1 > 0.0f) ? loss : -1.0f;
    }
}

extern "C" void kernel_launch(void* const* d_in, const int* in_sizes, int n_in,
                              void* d_out, int out_size, void* d_ws, size_t ws_size,
                              hipStream_t stream)
{
    (void)in_sizes; (void)n_in; (void)out_size; (void)ws_size;
    const float* y_pred = (const float*)d_in[0];
    const float* y_true = (const float*)d_in[1];
    const int*   bg     = (const int*)d_in[2];
    float* partials = (float*)d_ws;   // NPLANES * SPLIT * 3 floats = 32256 B

    dim3 grid(NPLANES, SPLIT);
    dice_partials_kernel<<<grid, THREADS, 0, stream>>>(y_pred, y_true, partials);
    dice_finalize_kernel<<<1, 384, 0, stream>>>(partials, bg, (float*)d_out);
}